// FromCodesModule_987842478744
// MI455X (gfx1250) — compile-verified
//
#include <hip/hip_runtime.h>

typedef __attribute__((ext_vector_type(2))) float v2f;
typedef __attribute__((ext_vector_type(8))) float v8f;

#define N_CB     14
#define VOCAB    1024
#define CB_DIM   8
#define OUT_CH   1024
#define TT       4096
#define KDIM     112           // N_CB * CB_DIM
#define KSTEPS   28            // KDIM / 4 (one v_wmma_f32_16x16x4_f32 per step)
#define KROW     116           // LDS row pitch (floats): l*52 mod 64 permutation -> conflict-free
#define M_CHUNK  128           // positions per workgroup
#define THREADS  256           // 8 wave32 waves

__global__ __launch_bounds__(THREADS) void fromcodes_gemm_kernel(
    const int*   __restrict__ codes,      // (B, 14, 4096) int32
    const float* __restrict__ codebooks,  // (14, 1024, 8)
    const float* __restrict__ proj_w,     // (14, 1024, 8)
    const float* __restrict__ proj_b,     // (14, 1024)
    float*       __restrict__ out)        // (B, 1024, 4096)
{
    __shared__ __align__(16) float zbuf[M_CHUNK * KROW];  // gathered Z (B operand) [pos][kk]
    __shared__ float bias[OUT_CH];                        // sum_k proj_b[k][o]

    const int tid    = threadIdx.x;
    const int m_base = blockIdx.x * M_CHUNK;     // global position index (b*T + t)
    const int b      = m_base / TT;              // 128 | 4096, so chunk stays in one b
    const int t_base = m_base - b * TT;

    // ---- phase 1a: summed bias into LDS -------------------------------------
    for (int o = tid; o < OUT_CH; o += THREADS) {
        float s = 0.f;
        #pragma unroll
        for (int k = 0; k < N_CB; ++k) s += proj_b[k * OUT_CH + o];
        bias[o] = s;
    }

    // ---- phase 1b: gather embeddings (128 pos x 14 codebooks x 32B) ---------
    for (int task = tid; task < M_CHUNK * N_CB; task += THREADS) {
        const int p = task / N_CB;
        const int k = task - p * N_CB;
        const int code = codes[(b * N_CB + k) * TT + t_base + p];
        const float* src = codebooks + ((size_t)(k * VOCAB + code)) * CB_DIM;
        const float4 lo = *(const float4*)(src);
        const float4 hi = *(const float4*)(src + 4);
        float* dst = &zbuf[p * KROW + k * CB_DIM];
        *(float4*)(dst)     = lo;
        *(float4*)(dst + 4) = hi;
    }
    __syncthreads();

    const int wave  = tid >> 5;     // wave32
    const int lane  = tid & 31;
    const int lh    = lane & 15;
    const int hi    = lane >> 4;    // hi half-wave handles K+2 / M+8 per WMMA layout

    // ---- phase 2: W-stationary WMMA GEMM ------------------------------------
    // Output tile: M = o (16), N = t (16). Each wave owns 8 o-tiles.
    for (int ot = 0; ot < 8; ++ot) {
        const int o0 = (wave * 8 + ot) * 16;

        // A-fragments (W): lane holds W[kk][o0+lh], kk pair = 4j + 2*hi + {0,1}.
        // proj_w D-dim contiguous -> one aligned 8B load per fragment. Held in 56 VGPRs.
        v2f wf[KSTEPS];
        #pragma unroll
        for (int j = 0; j < KSTEPS; ++j) {
            const int kk = 4 * j + 2 * hi;           // even -> never splits a D-row
            const int k  = kk >> 3;
            const int d  = kk & 7;
            wf[j] = *(const v2f*)(proj_w + ((size_t)(k * OUT_CH + o0 + lh)) * CB_DIM + d);
        }

        // bias per accumulator VGPR: VGPR v <-> row M = o0 + v + 8*hi
        float bv[8];
        #pragma unroll
        for (int v = 0; v < 8; ++v) bv[v] = bias[o0 + v + 8 * hi];

        for (int mt = 0; mt < M_CHUNK / 16; ++mt) {
            const int p0 = mt * 16;
            v8f c;
            #pragma unroll
            for (int v = 0; v < 8; ++v) c[v] = bv[v];

            #pragma unroll
            for (int j = 0; j < KSTEPS; ++j) {
                const int kk = 4 * j + 2 * hi;
                // B-fragment (Z): lane holds Z[kk..kk+1][p0+lh] from LDS (conflict-free pitch)
                const v2f zf = *(const v2f*)&zbuf[(p0 + lh) * KROW + kk];
                c = __builtin_amdgcn_wmma_f32_16x16x4_f32(
                        false, wf[j], false, zf, (short)0, c, false, false);
            }

            // store: VGPR v -> out[b][o0+v+8*hi][t_base+p0+lh]; lanes give 2x64B segments
            float* ob = out + (size_t)b * OUT_CH * TT + (size_t)(t_base + p0 + lh);
            #pragma unroll
            for (int v = 0; v < 8; ++v) {
                ob[(size_t)(o0 + v + 8 * hi) * TT] = c[v];
            }
        }
    }
}

extern "C" void kernel_launch(void* const* d_in, const int* in_sizes, int n_in,
                              void* d_out, int out_size, void* d_ws, size_t ws_size,
                              hipStream_t stream) {
    (void)n_in; (void)d_ws; (void)ws_size; (void)out_size;
    const int*   codes     = (const int*)d_in[0];
    const float* codebooks = (const float*)d_in[1];
    const float* proj_w    = (const float*)d_in[2];
    const float* proj_b    = (const float*)d_in[3];
    float*       out       = (float*)d_out;

    const int total_pos = in_sizes[0] / N_CB;         // B * T
    const int grid      = total_pos / M_CHUNK;        // 512 for B=16, T=4096

    hipLaunchKernelGGL(fromcodes_gemm_kernel, dim3(grid), dim3(THREADS), 0, stream,
                       codes, codebooks, proj_w, proj_b, out);
}